// ContextualRoutingForest_51565377356120
// MI455X (gfx1250) — compile-verified
//
#include <hip/hip_runtime.h>
#include <hip/hip_bf16.h>
#include <math.h>

// ---------------------------------------------------------------------------
// ContextualRoutingForest on MI455X (gfx1250), fp32 WMMA path.
// B=8, S=512, D=1024, O=1024, T=12, DEP=3, L=8 ; M = B*S = 4096
// ---------------------------------------------------------------------------

typedef __attribute__((ext_vector_type(2))) float v2f;
typedef __attribute__((ext_vector_type(8))) float v8f;
typedef __attribute__((ext_vector_type(4))) int   v4i;

#define M_ROWS 4096
#define D_DIM  1024
#define O_DIM  1024
#define T_NUM  12
#define KL     96          // T * L

#define TILE_M 128
#define TILE_N 128
#define KCH    32
#define LDA_S   36         // A tile row stride (floats): 16B aligned, conflict-free
#define LDB_GRP 288        // B tile K-pair-group stride (floats): 288%64==32 -> half-waves
                           // hit disjoint bank halves; 288*4 bytes is 16B aligned

// --- CDNA5 async global->LDS copy (ASYNCcnt path), with fallback ------------
#if __has_builtin(__builtin_amdgcn_global_load_async_to_lds_b128) && \
    __has_builtin(__builtin_amdgcn_s_wait_asynccnt)
#define USE_ASYNC_LDS 1
#else
#define USE_ASYNC_LDS 0
#endif

#if USE_ASYNC_LDS
typedef __attribute__((address_space(1))) v4i* v4i_gp;   // global int4*
typedef __attribute__((address_space(3))) v4i* v4i_lp;   // LDS    int4*
// one 16-byte async copy; OFF (bytes, imm) is added to BOTH memaddr and dsaddr
#define ASYNC_B128(gptr, lptr, OFF)                                            \
    __builtin_amdgcn_global_load_async_to_lds_b128(                            \
        (v4i_gp)(gptr), (v4i_lp)(lptr), (OFF), 0)
#define ASYNC_WAIT() __builtin_amdgcn_s_wait_asynccnt(0)
#endif

// ---------------------------------------------------------------------------
// 1) Causal EMA scan:  ema[s] = 0.1 * cumsum(x[s-1]*0.9^s) / (0.9^s + 1e-8)
//    One thread per (b,d) column, sequential over S (coalesced in d).
// ---------------------------------------------------------------------------
__global__ __launch_bounds__(256)
void ema_scan_kernel(const float* __restrict__ x, float* __restrict__ ema)
{
    int idx = blockIdx.x * 256 + threadIdx.x;        // 0 .. B*D-1  (8192)
    int b = idx >> 10;
    int d = idx & 1023;
    const float* xp = x   + (size_t)b * 512 * 1024 + d;
    float*       ep = ema + (size_t)b * 512 * 1024 + d;
    float c = 0.f;        // cumsum
    float p = 1.f;        // 0.9^s
    float prev = 0.f;     // x[s-1]
    for (int s = 0; s < 512; ++s) {
        c += prev * p;
        ep[(size_t)s * 1024] = 0.1f * c / (p + 1e-8f);
        prev = xp[(size_t)s * 1024];
        p *= 0.9f;
    }
}

// ---------------------------------------------------------------------------
// 2/4) fp32 WMMA GEMM:  C[M,N] = A[M,K] @ W[K,N]  (+ xadd + bias, optional)
//     Block tile 128x128, 8 waves; each wave owns a 16x128 strip with eight
//     16x16 f32 accumulators driven by V_WMMA_F32_16X16X4_F32.
//     A tile staged via GLOBAL_LOAD_ASYNC_TO_LDS_B128 (no VGPR round-trip);
//     B tile staged through VGPRs (K-pair zip) so every hot-loop fragment is
//     a single 8-byte ds_load.
// ---------------------------------------------------------------------------
__global__ __launch_bounds__(256)
void wmma_gemm_f32_kernel(const float* __restrict__ A, const float* __restrict__ W,
                          const float* __restrict__ xadd, const float* __restrict__ bias,
                          float* __restrict__ C, int M, int N, int K)
{
    __shared__ float lA[TILE_M * LDA_S];        // 128*36*4  = 18 KB
    __shared__ float lB[(KCH / 2) * LDB_GRP];   // 16*288*4  = 18 KB

    const int tid   = threadIdx.x;
    const int lane  = tid & 31;
    const int wv    = tid >> 5;
    const int m0    = blockIdx.y * TILE_M;
    const int n0    = blockIdx.x * TILE_N;
    const int mrow  = lane & 15;           // M index of A frag; N index of B and C frags
    const int khalf = lane >> 4;           // which K-pair this half-wave holds

    v8f acc[8] = {};

    for (int kc = 0; kc < K; kc += KCH) {
        // --- stage A tile: 128 rows x 32 cols (64 contiguous bytes / thread) ---
        {
            int r  = tid >> 1;
            int cb = (tid & 1) * 16;
            const float* ga = A + (size_t)(m0 + r) * K + kc + cb;
            float*       la = lA + r * LDA_S + cb;
#if USE_ASYNC_LDS
            ASYNC_B128(ga, la, 0);
            ASYNC_B128(ga, la, 16);
            ASYNC_B128(ga, la, 32);
            ASYNC_B128(ga, la, 48);
#else
            const float4* src = (const float4*)ga;
            float4*       dst = (float4*)la;
            dst[0] = src[0]; dst[1] = src[1]; dst[2] = src[2]; dst[3] = src[3];
#endif
        }
        // --- stage W tile: 32 rows x 128 cols, K-pair interleaved (zip in VGPRs) ---
        // unit = 2 rows x 4 cols; 512 units; thread u and u+256.
        #pragma unroll
        for (int j = 0; j < 2; ++j) {
            int u  = tid + j * 256;
            int kp = u >> 5;               // K-pair group 0..15  (k = 2*kp)
            int nu = u & 31;               // column group, n = 4*nu
            const float* g0 = W + (size_t)(kc + kp * 2 + 0) * N + n0 + nu * 4;
            const float* g1 = W + (size_t)(kc + kp * 2 + 1) * N + n0 + nu * 4;
            float4 a4 = *(const float4*)g0;
            float4 b4 = *(const float4*)g1;
            float* dst = lB + kp * LDB_GRP + nu * 8;
            float4 z0 = { a4.x, b4.x, a4.y, b4.y };
            float4 z1 = { a4.z, b4.z, a4.w, b4.w };
            ((float4*)dst)[0] = z0;
            ((float4*)dst)[1] = z1;
        }
#if USE_ASYNC_LDS
        ASYNC_WAIT();
#endif
        __syncthreads();

        const float* aBase = lA + (wv * 16 + mrow) * LDA_S;
        #pragma unroll
        for (int kk = 0; kk < KCH / 4; ++kk) {          // K-step of 4
            v2f a = *(const v2f*)(aBase + kk * 4 + khalf * 2);
            const float* bB = lB + (kk * 2 + khalf) * LDB_GRP + mrow * 2;
            #pragma unroll
            for (int nt = 0; nt < 8; ++nt) {
                v2f b = *(const v2f*)(bB + nt * 32);
                acc[nt] = __builtin_amdgcn_wmma_f32_16x16x4_f32(
                    false, a, false, b, (short)0, acc[nt], false, false);
            }
        }
        __syncthreads();
    }

    // --- epilogue: C frag layout  M = v + 8*khalf (+wave strip), N = lane&15 ---
    const int row0 = m0 + wv * 16 + khalf * 8;
    #pragma unroll
    for (int nt = 0; nt < 8; ++nt) {
        int col  = n0 + nt * 16 + mrow;
        float bv = bias ? bias[col] : 0.f;
        #pragma unroll
        for (int v = 0; v < 8; ++v) {
            int row = row0 + v;
            float r = acc[nt][v] + bv;
            if (xadd) r += xadd[(size_t)row * N + col];
            C[(size_t)row * N + col] = r;
        }
    }
}

// ---------------------------------------------------------------------------
// 3) Per-row routing: 36 decision dots + 12 gate dots, temperatures, sigmoid,
//    Kronecker leaf probs, gate softmax  ->  wlp[row, 96] = w_t * leafprob_tl
// ---------------------------------------------------------------------------
__global__ __launch_bounds__(256)
void decide_kernel(const float* __restrict__ routing, const float* __restrict__ x,
                   const float* __restrict__ dW, const float* __restrict__ db,
                   const float* __restrict__ gW, const float* __restrict__ gb,
                   const float* __restrict__ ntl, float* __restrict__ wlp)
{
    __shared__ float R[D_DIM];
    __shared__ float X[D_DIM];
    __shared__ float dec48[48];
    __shared__ float sig36[36];
    __shared__ float w12[12];

    const int row = blockIdx.x;
    const int tid = threadIdx.x;
    const int lane = tid & 31;
    const int wv   = tid >> 5;

#if USE_ASYNC_LDS
    ASYNC_B128(routing + (size_t)row * D_DIM + tid * 4, R + tid * 4, 0);
    ASYNC_B128(x       + (size_t)row * D_DIM + tid * 4, X + tid * 4, 0);
    ASYNC_WAIT();
#else
    ((float4*)R)[tid] = ((const float4*)(routing + (size_t)row * D_DIM))[tid];
    ((float4*)X)[tid] = ((const float4*)(x       + (size_t)row * D_DIM))[tid];
#endif
    __syncthreads();

    // 48 dot products of length 1024; wave w handles dots w, w+8, ...
    for (int j = wv; j < 48; j += 8) {
        float acc = 0.f;
        if (j < 36) {
            const float* wr = dW + (size_t)j * D_DIM;   // dW flat [36,1024]
            for (int d = lane; d < D_DIM; d += 32) acc += R[d] * wr[d];
        } else {
            int t = j - 36;
            for (int d = lane; d < D_DIM; d += 32) acc += X[d] * gW[(size_t)d * T_NUM + t];
        }
        #pragma unroll
        for (int off = 16; off > 0; off >>= 1) acc += __shfl_xor(acc, off, 32);
        if (lane == 0) dec48[j] = acc + (j < 36 ? db[j] : gb[j - 36]);
    }
    __syncthreads();

    if (tid < 36) {
        float z  = ntl[tid] + 0.5413f;
        float sp = (z > 20.f) ? z : log1pf(expf(z));    // softplus, TEMP = 1
        sig36[tid] = 1.f / (1.f + expf(-dec48[tid] / sp));
    }
    if (tid == 0) {
        float mx = dec48[36];
        for (int t = 1; t < T_NUM; ++t) mx = fmaxf(mx, dec48[36 + t]);
        float sm = 0.f;
        for (int t = 0; t < T_NUM; ++t) { float e = expf(dec48[36 + t] - mx); w12[t] = e; sm += e; }
        float inv = 1.f / sm;
        for (int t = 0; t < T_NUM; ++t) w12[t] *= inv;
    }
    __syncthreads();

    if (tid < KL) {
        int t = tid >> 3, l = tid & 7;
        float s0 = sig36[t * 3 + 0], s1 = sig36[t * 3 + 1], s2 = sig36[t * 3 + 2];
        float p = ((l & 4) ? 1.f - s0 : s0)
                * ((l & 2) ? 1.f - s1 : s1)
                * ((l & 1) ? 1.f - s2 : s2);
        wlp[(size_t)row * KL + tid] = w12[t] * p;
    }
}

// ---------------------------------------------------------------------------
// 5) LayerNorm over O=1024 (biased variance, eps=1e-5)
// ---------------------------------------------------------------------------
__global__ __launch_bounds__(256)
void layernorm_kernel(const float* __restrict__ src, const float* __restrict__ gamma,
                      const float* __restrict__ beta, float* __restrict__ out)
{
    __shared__ float ws1[8], ws2[8];
    const int row = blockIdx.x;
    const int tid = threadIdx.x;
    const int lane = tid & 31, wv = tid >> 5;

    float4 v = ((const float4*)(src + (size_t)row * O_DIM))[tid];
    float s  = v.x + v.y + v.z + v.w;
    float s2 = v.x * v.x + v.y * v.y + v.z * v.z + v.w * v.w;
    #pragma unroll
    for (int off = 16; off > 0; off >>= 1) {
        s  += __shfl_xor(s,  off, 32);
        s2 += __shfl_xor(s2, off, 32);
    }
    if (lane == 0) { ws1[wv] = s; ws2[wv] = s2; }
    __syncthreads();
    if (tid == 0) {
        float a = 0.f, b = 0.f;
        for (int i = 0; i < 8; ++i) { a += ws1[i]; b += ws2[i]; }
        float mu  = a * (1.f / O_DIM);
        float var = b * (1.f / O_DIM) - mu * mu;
        ws1[0] = mu;
        ws2[0] = rsqrtf(var + 1e-5f);
    }
    __syncthreads();
    float mu = ws1[0], inv = ws2[0];
    float4 g  = ((const float4*)gamma)[tid];
    float4 be = ((const float4*)beta)[tid];
    float4 o;
    o.x = (v.x - mu) * inv * g.x + be.x;
    o.y = (v.y - mu) * inv * g.y + be.y;
    o.z = (v.z - mu) * inv * g.z + be.z;
    o.w = (v.w - mu) * inv * g.w + be.w;
    ((float4*)(out + (size_t)row * O_DIM))[tid] = o;
}

// ---------------------------------------------------------------------------
// Launch
// ---------------------------------------------------------------------------
extern "C" void kernel_launch(void* const* d_in, const int* in_sizes, int n_in,
                              void* d_out, int out_size, void* d_ws, size_t ws_size,
                              hipStream_t stream)
{
    const float* x        = (const float*)d_in[0];
    const float* cW       = (const float*)d_in[1];
    const float* cb       = (const float*)d_in[2];
    const float* dW       = (const float*)d_in[3];
    const float* db       = (const float*)d_in[4];
    const float* leaf_out = (const float*)d_in[5];
    const float* gW       = (const float*)d_in[6];
    const float* gb       = (const float*)d_in[7];
    const float* ntl      = (const float*)d_in[8];
    const float* gamma    = (const float*)d_in[9];
    const float* beta     = (const float*)d_in[10];
    float* out = (float*)d_out;

    // Workspace layout (~35 MB): ema is dead after the routing GEMM, reuse it
    // as the pre-LayerNorm output buffer.
    float* ema     = (float*)d_ws;                         // [4096,1024]
    float* routing = ema     + (size_t)M_ROWS * D_DIM;     // [4096,1024]
    float* wlp     = routing + (size_t)M_ROWS * D_DIM;     // [4096,  96]
    float* out0    = ema;                                  // alias  [4096,1024]

    // 1) EMA scan (B*D = 8192 columns)
    ema_scan_kernel<<<32, 256, 0, stream>>>(x, ema);

    // 2) routing = x + ema @ cW + cb     (M=4096, N=1024, K=1024)
    wmma_gemm_f32_kernel<<<dim3(D_DIM / TILE_N, M_ROWS / TILE_M), 256, 0, stream>>>(
        ema, cW, x, cb, routing, M_ROWS, D_DIM, D_DIM);

    // 3) decisions, temperatures, leaf probs, gate softmax -> wlp [4096,96]
    decide_kernel<<<M_ROWS, 256, 0, stream>>>(routing, x, dW, db, gW, gb, ntl, wlp);

    // 4) out0 = wlp @ leaf_out           (M=4096, N=1024, K=96)
    wmma_gemm_f32_kernel<<<dim3(O_DIM / TILE_N, M_ROWS / TILE_M), 256, 0, stream>>>(
        wlp, leaf_out, nullptr, nullptr, out0, M_ROWS, O_DIM, KL);

    // 5) LayerNorm
    layernorm_kernel<<<M_ROWS, 256, 0, stream>>>(out0, gamma, beta, out);
}